// MyLoss_62929860821445
// MI455X (gfx1250) — compile-verified
//
#include <hip/hip_runtime.h>
#include <math.h>

// Problem constants (from reference): feature [16,64,64,64], 5x5 patches stride 2.
#define B_   16
#define C_   64
#define H_   64
#define W_   64
#define P_   30            // patches per spatial dim: (64-5)/2+1
#define L_   900           // P_*P_
#define D_   1600          // C_*25
#define NG_  57            // ceil(L_/16) patch groups per batch
#define EPSN 1e-6f

typedef __attribute__((ext_vector_type(2))) float v2f;
typedef __attribute__((ext_vector_type(8))) float v8f;

// ---------------- init: reset atomic-min slots (graph-replay safe) -------------
__global__ void k_init(unsigned* __restrict__ minu) {
    if (threadIdx.x < B_) minu[threadIdx.x] = 0xFFFFFFFFu;
}

// ---------------- 1) per-patch inverse norms ----------------------------------
__global__ void k_norm(const float* __restrict__ feat, float* __restrict__ invn) {
    int idx = blockIdx.x * blockDim.x + threadIdx.x;
    if (idx >= B_ * L_) return;
    int b = idx / L_, l = idx % L_;
    int ph = l / P_, pw = l % P_;
    const float* fb = feat + (size_t)b * C_ * H_ * W_ + (size_t)(2 * ph) * W_ + 2 * pw;
    float s = 0.f;
    for (int c = 0; c < C_; ++c) {
        const float* fc = fb + (size_t)c * H_ * W_;
        #pragma unroll
        for (int dy = 0; dy < 5; ++dy)
            #pragma unroll
            for (int dx = 0; dx < 5; ++dx) {
                float v = fc[dy * W_ + dx];
                s += v * v;
            }
    }
    invn[idx] = 1.f / fmaxf(sqrtf(s), EPSN);
}

// ---------------- 2) g[b,d] = sum_l fn[b,l,d]  --------------------------------
__global__ void k_gsum(const float* __restrict__ feat, const float* __restrict__ invn,
                       float* __restrict__ g) {
    int idx = blockIdx.x * blockDim.x + threadIdx.x;
    if (idx >= B_ * D_) return;
    int b = idx / D_, d = idx % D_;
    int c = d / 25, r = d % 25, dy = r / 5, dx = r % 5;
    const float* fb = feat + ((size_t)(b * C_ + c) * H_ + dy) * W_ + dx;
    const float* nv = invn + (size_t)b * L_;
    float s = 0.f;
    for (int ph = 0; ph < P_; ++ph) {
        const float* frow = fb + (size_t)(2 * ph) * W_;
        for (int pw = 0; pw < P_; ++pw)
            s += frow[2 * pw] * nv[ph * P_ + pw];
    }
    g[idx] = s;
}

// ---------------- 3) dots fn[l].g via V_WMMA_F32_16X16X4_F32, min-reduce -------
// One wave (32 lanes) per 16-patch group. A = 16x4 tile of fn (rows = patches),
// B = g chunk replicated across the 16 columns, C/D accumulates over 400 chunks.
__global__ void __launch_bounds__(32)
k_dotmin(const float* __restrict__ feat, const float* __restrict__ invn,
         const float* __restrict__ g, unsigned* __restrict__ minu) {
    int b    = blockIdx.x / NG_;
    int grp  = blockIdx.x % NG_;
    int lane = threadIdx.x;          // 0..31, wave32
    int m    = lane & 15;            // A-matrix row (patch within group)
    int half = lane >> 4;            // 0: K=0,1  1: K=2,3 (32-bit A layout)

    int patch  = grp * 16 + m;
    int pclamp = (patch < L_) ? patch : (L_ - 1);
    int ph = pclamp / P_, pw = pclamp % P_;
    const float* fb = feat + ((size_t)(b * C_) * H_ + 2 * ph) * W_ + 2 * pw;
    // Zero out rows for tail padding -> their dots are 0 and masked from the min.
    float inm = (patch < L_) ? invn[(size_t)b * L_ + pclamp] : 0.f;
    const float* gb = g + (size_t)b * D_;

    v8f acc = {};
    for (int t = 0; t < D_ / 4; ++t) {
        int da = 4 * t + 2 * half;   // this lane's first K element
        int db = da + 1;
        int c0 = da / 25, r0 = da % 25;
        int c1 = db / 25, r1 = db % 25;
        // fn[patch][da], fn[patch][db] gathered from the (never materialized) unfold
        float a0 = fb[((size_t)c0 * H_ + r0 / 5) * W_ + (r0 % 5)] * inm;
        float a1 = fb[((size_t)c1 * H_ + r1 / 5) * W_ + (r1 % 5)] * inm;
        v2f av; av.x = a0;     av.y = a1;
        v2f bv; bv.x = gb[da]; bv.y = gb[db];   // broadcast g across all 16 cols
        __builtin_prefetch(gb + da + 64, 0, 0); // global_prefetch_b8
        acc = __builtin_amdgcn_wmma_f32_16x16x4_f32(
            /*neg_a=*/false, av, /*neg_b=*/false, bv,
            /*c_mod=*/(short)0, acc, /*reuse_a=*/false, /*reuse_b=*/false);
    }

    // D layout: lanes 0-15 = cols 0-15 rows 0-7 (vgpr r = row r);
    // lanes 16-31 = rows 8-15. Column 0 lives in lanes 0 and 16.
    if ((lane & 15) == 0) {
        float mn = 3.4e38f;
        #pragma unroll
        for (int r = 0; r < 8; ++r) {
            int p = grp * 16 + half * 8 + r;
            if (p < L_) mn = fminf(mn, acc[r]);
        }
        unsigned bits = __float_as_uint(mn);
        // order-preserving map so unsigned atomicMin == float min
        unsigned key = (bits & 0x80000000u) ? ~bits : (bits ^ 0x80000000u);
        atomicMin(&minu[b], key);
    }
}

// ---------------- 4) final scalar loss ----------------------------------------
__global__ void k_final(const unsigned* __restrict__ minu, const int* __restrict__ label,
                        float* __restrict__ out) {
    if (threadIdx.x != 0 || blockIdx.x != 0) return;
    float fsum = 0.f, rsum = 0.f, fcnt = 0.f, rcnt = 0.f;
    for (int b = 0; b < B_; ++b) {
        unsigned key  = minu[b];
        unsigned bits = (key & 0x80000000u) ? (key ^ 0x80000000u) : ~key;
        float dmin = __uint_as_float(bits);
        float top1 = 1.f - dmin / (float)L_;   // max_l s[l] = 1 - min_l(fn.g)/L
        if (label[b] == 1) { fsum += top1; fcnt += 1.f; }
        else               { rsum += top1; rcnt += 1.f; }
    }
    float loss = 1.f - fsum / fcnt + rsum / rcnt;
    out[0] = fmaxf(loss, 0.f);
}

extern "C" void kernel_launch(void* const* d_in, const int* in_sizes, int n_in,
                              void* d_out, int out_size, void* d_ws, size_t ws_size,
                              hipStream_t stream) {
    const float* feat  = (const float*)d_in[0];
    const int*   label = (const int*)d_in[1];
    float*       out   = (float*)d_out;

    // Workspace layout: invn [B*L] f32 | g [B*D] f32 | minu [B] u32  (~160 KB)
    float*    invn = (float*)d_ws;
    float*    g    = invn + (size_t)B_ * L_;
    unsigned* minu = (unsigned*)(g + (size_t)B_ * D_);

    k_init  <<<1, 32, 0, stream>>>(minu);
    k_norm  <<<(B_ * L_ + 255) / 256, 256, 0, stream>>>(feat, invn);
    k_gsum  <<<(B_ * D_ + 255) / 256, 256, 0, stream>>>(feat, invn, g);
    k_dotmin<<<B_ * NG_, 32, 0, stream>>>(feat, invn, g, minu);
    k_final <<<1, 1, 0, stream>>>(minu, label, out);
}